// PerlinAttention_7842610283160
// MI455X (gfx1250) — compile-verified
//
#include <hip/hip_runtime.h>
#include <math.h>

// ---------------------------------------------------------------------------
// PerlinAttention for MI455X (gfx1250), f32 end-to-end via V_WMMA_F32_16X16X4_F32.
// Pipeline:
//   k_proj      : (x * d^-1/4) @ proj^T  -> pre-exp logits (M padded 266->272)
//   k_rowstats  : per-row max + diag
//   k_bhmax     : per-(b,h) max for key stabilizer
//   k_featexp   : performer feature map exp (+ per-row qp sum for eps term)
//   k_linattn   : chunked causal linear attention; S state in LDS; vc tile
//                 staged via GLOBAL_LOAD_ASYNC_TO_LDS_B128 (ASYNCcnt) and
//                 overlapped with the q@k^T WMMA phase
//   k_pred      : Linear+LN+GELU+Linear+softmax, WMMA GEMMs + LDS row ops
//   k_vbar      : fold bilinear interp weights into V ->  Vbar[L,D], svec[L]
//   k_final     : out = (est @ Vbar) / (est . svec + eps) + ctx   (WMMA)
// Workspace use: ~147 MB of d_ws (fits in the 192 MB L2).
// ---------------------------------------------------------------------------

constexpr int Bc  = 2, Hc = 12, Tt = 2048, Dd = 64;
constexpr int Mm  = 266;          // performer features
constexpr int Ll  = 128;          // predictor length
constexpr int BH  = Bc * Hc;      // 24
constexpr int BHT = BH * Tt;      // 49152 rows
constexpr int Mp  = 272;          // M padded to 17*16 for WMMA tiling
constexpr int CH  = 128;          // causal chunk
constexpr int NCH = Tt / CH;      // 16

typedef __attribute__((ext_vector_type(2))) float v2f;
typedef __attribute__((ext_vector_type(8))) float v8f;

__device__ __forceinline__ v8f wmma_f32(v2f a, v2f b, v8f c) {
  // (neg_a, A, neg_b, B, c_mod, C, reuse_a, reuse_b)
  return __builtin_amdgcn_wmma_f32_16x16x4_f32(false, a, false, b, (short)0, c,
                                               false, false);
}

__device__ __forceinline__ v8f vzero8() {
  v8f z = {0.f, 0.f, 0.f, 0.f, 0.f, 0.f, 0.f, 0.f};
  return z;
}

// async copy of 16B/lane from global (sgpr base + vgpr offset) into LDS.
__device__ __forceinline__ void async_g2l_b128(unsigned lds_byte_addr,
                                               unsigned g_byte_off,
                                               const void* g_base) {
  asm volatile("global_load_async_to_lds_b128 %0, %1, %2"
               :
               : "v"(lds_byte_addr), "v"(g_byte_off),
                 "s"((unsigned long long)(uintptr_t)g_base)
               : "memory");
}
__device__ __forceinline__ void wait_async0() {
  asm volatile("s_wait_asynccnt 0x0" ::: "memory");
}

// ---------------------------------------------------------------------------
// 1) dd[row, m] = sum_d (data[row,d] * 64^-0.25) * proj[m,d]
// One wave per 16x16 output tile, K = 64 -> 16 wmma steps.
// Padded output columns (m >= 266) get garbage (clamped proj row) -- they are
// never consumed: rowstats reads m<266, featexp overwrites them with zero.
__global__ void k_proj(const float* __restrict__ data,
                       const float* __restrict__ proj,
                       float* __restrict__ dd) {
  const int mtiles = Mp / 16;  // 17
  int gw = blockIdx.x * (blockDim.x >> 5) + (threadIdx.x >> 5);
  int rt = gw / mtiles, mt = gw % mtiles;
  if (rt >= BHT / 16) return;  // wave-uniform
  const int lane = threadIdx.x & 31, r = lane & 15, hi = lane >> 4;
  const int r0 = rt * 16, m0 = mt * 16;
  const float dn = 0.35355339059327373f;  // 64^-0.25
  const int mc = m0 + r;
  const int mcc = (mc < Mm) ? mc : (Mm - 1);  // clamp: no divergent loads
  v8f acc = vzero8();
#pragma unroll
  for (int k0 = 0; k0 < Dd; k0 += 4) {
    const float* ap = data + (size_t)(r0 + r) * Dd + k0 + hi * 2;
    v2f a; a.x = ap[0] * dn; a.y = ap[1] * dn;
    const float* bp = proj + (size_t)mcc * Dd + k0 + hi * 2;
    v2f b; b.x = bp[0]; b.y = bp[1];
    acc = wmma_f32(a, b, acc);
  }
#pragma unroll
  for (int e = 0; e < 8; ++e)
    dd[(size_t)(r0 + e + hi * 8) * Mp + m0 + r] = acc[e];
}

// ---------------------------------------------------------------------------
// 2) per-row: max over m of dd, and diag = 0.5*sum(data^2)*64^-0.5
__global__ void k_rowstats(const float* __restrict__ dd,
                           const float* __restrict__ data,
                           float* __restrict__ rowmax,
                           float* __restrict__ diag) {
  int row = blockIdx.x * (blockDim.x >> 5) + (threadIdx.x >> 5);
  int lane = threadIdx.x & 31;
  if (row >= BHT) return;
  float mx = -1e30f;
  for (int m = lane; m < Mm; m += 32) mx = fmaxf(mx, dd[(size_t)row * Mp + m]);
#pragma unroll
  for (int off = 16; off; off >>= 1) mx = fmaxf(mx, __shfl_xor(mx, off, 32));
  float s = 0.f;
  for (int d = lane; d < Dd; d += 32) {
    float x = data[(size_t)row * Dd + d];
    s += x * x;
  }
#pragma unroll
  for (int off = 16; off; off >>= 1) s += __shfl_xor(s, off, 32);
  if (lane == 0) { rowmax[row] = mx; diag[row] = 0.5f * s * 0.125f; }
}

// per-(b,h) max of rowmax_k
__global__ void k_bhmax(const float* __restrict__ rmk, float* __restrict__ stab) {
  __shared__ float red[256];
  int bh = blockIdx.x, t = threadIdx.x;
  float mx = -1e30f;
  for (int i = t; i < Tt; i += blockDim.x) mx = fmaxf(mx, rmk[(size_t)bh * Tt + i]);
  red[t] = mx; __syncthreads();
  for (int s = 128; s > 0; s >>= 1) {
    if (t < s) red[t] = fmaxf(red[t], red[t + s]);
    __syncthreads();
  }
  if (t == 0) stab[bh] = red[0];
}

// ---------------------------------------------------------------------------
// 3) feat = ratio*(exp(dd - diag - stab) + 1e-4), zero padded cols; qsum opt.
__global__ void k_featexp(float* __restrict__ dd, const float* __restrict__ diag,
                          const float* __restrict__ rowmax,
                          const float* __restrict__ stab_bh,
                          float* __restrict__ qsum, int is_query) {
  int row = blockIdx.x * (blockDim.x >> 5) + (threadIdx.x >> 5);
  int lane = threadIdx.x & 31;
  if (row >= BHT) return;
  float sub = diag[row] + (is_query ? rowmax[row] : stab_bh[row / Tt]);
  const float ratio = rsqrtf(266.0f);
  float ssum = 0.f;
  for (int m = lane; m < Mp; m += 32) {
    float v = (m < Mm) ? ratio * (expf(dd[(size_t)row * Mp + m] - sub) + 1e-4f)
                       : 0.f;
    dd[(size_t)row * Mp + m] = v;
    ssum += v;
  }
  if (qsum) {
#pragma unroll
    for (int off = 16; off; off >>= 1) ssum += __shfl_xor(ssum, off, 32);
    if (lane == 0) qsum[row] = ssum;
  }
}

// ---------------------------------------------------------------------------
// 4) chunked causal linear attention. One workgroup (8 waves) per (b,h).
//    LDS: S[Mp][Dd] + ksum[Mp] + Amask[8][16][CH] + dinv[8][16] + Vs[CH][Dd]
//    (~170 KB of the 320 KB WGP pool). vc tile arrives via async DMA that
//    overlaps the phase-A WMMA work.
__global__ void k_linattn(const float* __restrict__ qp,
                          const float* __restrict__ kp,
                          const float* __restrict__ v,
                          const float* __restrict__ qsum,
                          float* __restrict__ ctx) {
  extern __shared__ float sm[];
  float* S    = sm;                  // [Mp][Dd]
  float* ksum = S + Mp * Dd;         // [Mp]
  float* Abuf = ksum + Mp;           // [8][16][CH]
  float* dbuf = Abuf + 8 * 16 * CH;  // [8][16]
  float* Vs   = dbuf + 8 * 16;       // [CH][Dd]  staged v chunk
  const int bh = blockIdx.x;
  const int tid = threadIdx.x, wv = tid >> 5, lane = tid & 31;
  const int r = lane & 15, hi = lane >> 4;
  const float* qpb = qp + (size_t)bh * Tt * Mp;
  const float* kpb = kp + (size_t)bh * Tt * Mp;
  const float* vb  = v  + (size_t)bh * Tt * Dd;
  float* ctxb = ctx + (size_t)bh * Tt * Dd;
  float* Aw = Abuf + wv * 16 * CH;
  const unsigned vs_lds = (unsigned)(uintptr_t)Vs;  // low 32 bits = LDS offset

  for (int i = tid; i < Mp * Dd; i += blockDim.x) S[i] = 0.f;
  for (int i = tid; i < Mp; i += blockDim.x) ksum[i] = 0.f;
  __syncthreads();

  for (int c = 0; c < NCH; ++c) {
    const int t0 = c * CH;
    const int rb = t0 + wv * 16;  // this wave's 16-row q band

    // kick off async DMA of this chunk's v tile (32 KB) into LDS; the copy
    // runs while phase A does its q@k^T WMMA work.  8 x 16B per thread.
    {
      const void* gbase = (const void*)(vb + (size_t)t0 * Dd);
#pragma unroll
      for (int it = 0; it < (CH * Dd * 4) / (256 * 16); ++it) {
        unsigned boff = (unsigned)(tid * 16 + it * 4096);
        async_g2l_b128(vs_lds + boff, boff, gbase);
      }
    }
    if (c + 1 < NCH)  // warm L2/WGP$ for next chunk's K features
      __builtin_prefetch(kpb + (size_t)(t0 + CH + wv * 16) * Mp, 0, 0);

    // ---- Phase A: A = q_band (16 x Mp) @ kc^T (Mp x 128) ----
    v8f acc[8];
#pragma unroll
    for (int j = 0; j < 8; ++j) acc[j] = vzero8();
    for (int m0 = 0; m0 < Mp; m0 += 4) {
      const float* ap = qpb + (size_t)(rb + r) * Mp + m0 + hi * 2;
      v2f a; a.x = ap[0]; a.y = ap[1];
#pragma unroll
      for (int j = 0; j < 8; ++j) {
        const float* bp = kpb + (size_t)(t0 + j * 16 + r) * Mp + m0 + hi * 2;
        v2f b; b.x = bp[0]; b.y = bp[1];
        acc[j] = wmma_f32(a, b, acc[j]);
      }
    }
    // causal mask within chunk, stash to LDS (layout change for next GEMM)
#pragma unroll
    for (int j = 0; j < 8; ++j)
#pragma unroll
      for (int e = 0; e < 8; ++e) {
        int gi = wv * 16 + e + hi * 8;  // row in chunk
        int gj = j * 16 + r;            // col in chunk
        Aw[(e + hi * 8) * CH + gj] = (gj <= gi) ? acc[j][e] : 0.f;
      }
    wait_async0();    // our slice of the v tile has landed in LDS
    __syncthreads();  // ... and everyone else's

    // ---- Phase B: out = A_mask @ vc + q_band @ S ----
    v8f oacc[4];
#pragma unroll
    for (int dt = 0; dt < 4; ++dt) oacc[dt] = vzero8();
    for (int k0 = 0; k0 < CH; k0 += 4) {
      v2f a;
      a.x = Aw[r * CH + k0 + hi * 2];
      a.y = Aw[r * CH + k0 + hi * 2 + 1];
#pragma unroll
      for (int dt = 0; dt < 4; ++dt) {
        const float* bp = Vs + (size_t)(k0 + hi * 2) * Dd + dt * 16 + r;
        v2f b; b.x = bp[0]; b.y = bp[Dd];
        oacc[dt] = wmma_f32(a, b, oacc[dt]);
      }
    }
    for (int m0 = 0; m0 < Mp; m0 += 4) {
      const float* ap = qpb + (size_t)(rb + r) * Mp + m0 + hi * 2;
      v2f a; a.x = ap[0]; a.y = ap[1];
#pragma unroll
      for (int dt = 0; dt < 4; ++dt) {
        const float* bp = S + (size_t)(m0 + hi * 2) * Dd + dt * 16 + r;
        v2f b; b.x = bp[0]; b.y = bp[Dd];
        oacc[dt] = wmma_f32(a, b, oacc[dt]);
      }
    }
    // denom_i = rowsum(A_mask)_i + q_i . ksum_prev + eps * qsum_i
    if (lane < 16) {
      float srow = 0.f;
      for (int j2 = 0; j2 < CH; ++j2) srow += Aw[lane * CH + j2];
      float dotk = 0.f;
      const float* qrow = qpb + (size_t)(rb + lane) * Mp;
      for (int m = 0; m < Mm; ++m) dotk += qrow[m] * ksum[m];
      float den = srow + dotk + 1e-6f * qsum[(size_t)bh * Tt + rb + lane];
      dbuf[wv * 16 + lane] = 1.0f / den;
    }
#pragma unroll
    for (int dt = 0; dt < 4; ++dt)
#pragma unroll
      for (int e = 0; e < 8; ++e)
        ctxb[(size_t)(rb + e + hi * 8) * Dd + dt * 16 + r] =
            oacc[dt][e] * dbuf[wv * 16 + e + hi * 8];
    __syncthreads();  // all waves done reading S / ksum for this chunk

    // ---- Phase C: S += kc^T (Mp x 128) @ vc (128 x 64) ----
    for (int tile = wv; tile < (Mp / 16) * 4; tile += 8) {
      int mt = tile >> 2, dt = tile & 3;
      v8f sc;
#pragma unroll
      for (int e = 0; e < 8; ++e)
        sc[e] = S[(size_t)(mt * 16 + e + hi * 8) * Dd + dt * 16 + r];
      for (int k0 = 0; k0 < CH; k0 += 4) {
        const float* apk = kpb + (size_t)(t0 + k0 + hi * 2) * Mp + mt * 16 + r;
        v2f a; a.x = apk[0]; a.y = apk[Mp];
        const float* bpv = Vs + (size_t)(k0 + hi * 2) * Dd + dt * 16 + r;
        v2f b; b.x = bpv[0]; b.y = bpv[Dd];
        sc = wmma_f32(a, b, sc);
      }
#pragma unroll
      for (int e = 0; e < 8; ++e)
        S[(size_t)(mt * 16 + e + hi * 8) * Dd + dt * 16 + r] = sc[e];
    }
    // ksum += column sums of this chunk's k features
    for (int m = tid; m < Mp; m += blockDim.x) {
      float s2 = 0.f;
      for (int j2 = 0; j2 < CH; ++j2) s2 += kpb[(size_t)(t0 + j2) * Mp + m];
      ksum[m] += s2;
    }
    __syncthreads();
  }
}

// ---------------------------------------------------------------------------
// 5) predictor: est = softmax(GELU(LN([ctx,v,q] @ Wenc + benc)) @ Wdec + bdec)
//    One wave per 16-row band; per-wave 16x128 LDS tile for LN/softmax.
__global__ void k_pred(const float* __restrict__ ctx, const float* __restrict__ v,
                       const float* __restrict__ q, const float* __restrict__ Wenc,
                       const float* __restrict__ benc, const float* __restrict__ lng,
                       const float* __restrict__ lnb, const float* __restrict__ Wdec,
                       const float* __restrict__ bdec, float* __restrict__ est) {
  extern __shared__ float sm[];
  const int tid = threadIdx.x, wv = tid >> 5, lane = tid & 31;
  const int r = lane & 15, hi = lane >> 4;
  float* Hw = sm + wv * 16 * Ll;
  const int band = blockIdx.x * (blockDim.x >> 5) + wv;
  const int r0 = band * 16;

  // GEMM1: y = enc_in @ Wenc  (K = 192, concat selects source tensor per k0)
  v8f acc[8];
#pragma unroll
  for (int nt = 0; nt < 8; ++nt) acc[nt] = vzero8();
  for (int k0 = 0; k0 < 3 * Dd; k0 += 4) {
    const float* src = (k0 < Dd) ? ctx : (k0 < 2 * Dd) ? v : q;
    int kk = (k0 < Dd) ? k0 : (k0 < 2 * Dd) ? (k0 - Dd) : (k0 - 2 * Dd);
    const float* ap = src + (size_t)(r0 + r) * Dd + kk + hi * 2;
    v2f a; a.x = ap[0]; a.y = ap[1];
#pragma unroll
    for (int nt = 0; nt < 8; ++nt) {
      const float* bp = Wenc + (size_t)(k0 + hi * 2) * Ll + nt * 16 + r;
      v2f b; b.x = bp[0]; b.y = bp[Ll];
      acc[nt] = wmma_f32(a, b, acc[nt]);
    }
  }
#pragma unroll
  for (int nt = 0; nt < 8; ++nt)
#pragma unroll
    for (int e = 0; e < 8; ++e) {
      int col = nt * 16 + r;
      Hw[(e + hi * 8) * Ll + col] = acc[nt][e] + benc[col];
    }
  // LN + exact GELU, one row per lane (lanes 0..15)
  if (lane < 16) {
    float mu = 0.f;
    for (int c2 = 0; c2 < Ll; ++c2) mu += Hw[lane * Ll + c2];
    mu *= (1.0f / Ll);
    float var = 0.f;
    for (int c2 = 0; c2 < Ll; ++c2) {
      float d = Hw[lane * Ll + c2] - mu;
      var += d * d;
    }
    var *= (1.0f / Ll);
    float is = rsqrtf(var + 1e-5f);
    for (int c2 = 0; c2 < Ll; ++c2) {
      float xh = (Hw[lane * Ll + c2] - mu) * is * lng[c2] + lnb[c2];
      Hw[lane * Ll + c2] = 0.5f * xh * (1.0f + erff(xh * 0.70710678118654752f));
    }
  }
  // GEMM2: z = H @ Wdec (K = 128), A-frags from LDS
#pragma unroll
  for (int nt = 0; nt < 8; ++nt) acc[nt] = vzero8();
  for (int k0 = 0; k0 < Ll; k0 += 4) {
    v2f a;
    a.x = Hw[r * Ll + k0 + hi * 2];
    a.y = Hw[r * Ll + k0 + hi * 2 + 1];
#pragma unroll
    for (int nt = 0; nt < 8; ++nt) {
      const float* bp = Wdec + (size_t)(k0 + hi * 2) * Ll + nt * 16 + r;
      v2f b; b.x = bp[0]; b.y = bp[Ll];
      acc[nt] = wmma_f32(a, b, acc[nt]);
    }
  }
#pragma unroll
  for (int nt = 0; nt < 8; ++nt)
#pragma unroll
    for (int e = 0; e < 8; ++e) {
      int col = nt * 16 + r;
      Hw[(e + hi * 8) * Ll + col] = acc[nt][e] + bdec[col];
    }
  // softmax per row, write est
  if (lane < 16) {
    float mx = -1e30f;
    for (int c2 = 0; c2 < Ll; ++c2) mx = fmaxf(mx, Hw[lane * Ll + c2]);
    float sum = 0.f;
    for (int c2 = 0; c2 < Ll; ++c2) {
      float ex = expf(Hw[lane * Ll + c2] - mx);
      Hw[lane * Ll + c2] = ex;
      sum += ex;
    }
    float inv = 1.0f / sum;
    for (int c2 = 0; c2 < Ll; ++c2)
      est[(size_t)(r0 + lane) * Ll + c2] = Hw[lane * Ll + c2] * inv;
  }
}

// ---------------------------------------------------------------------------
// 6) fold interp weights into V: Vbar[bh,l,d] = sum_t w(t->l)*v[bh,t,d],
//    svec[l] = sum_t w(t->l). Each t only hits l in [16l-8, 16l+24).
__global__ void k_vbar(const float* __restrict__ v, float* __restrict__ Vbar,
                       float* __restrict__ svec) {
  const int bh = blockIdx.y, l = blockIdx.x, d = threadIdx.x;  // 64 threads
  const float scale = (float)Ll / (float)Tt;  // 1/16
  float accv = 0.f, accs = 0.f;
  int tlo = l * 16 - 8; if (tlo < 0) tlo = 0;
  int thi = l * 16 + 24; if (thi > Tt) thi = Tt;
  for (int t = tlo; t < thi; ++t) {
    float pos = ((float)t + 0.5f) * scale - 0.5f;
    float lof = floorf(pos);
    float w = pos - lof;
    int lo = (int)lof;
    int lo_i = lo < 0 ? 0 : (lo > Ll - 1 ? Ll - 1 : lo);
    int hi_i = lo + 1 < 0 ? 0 : (lo + 1 > Ll - 1 ? Ll - 1 : lo + 1);
    float wt = ((lo_i == l) ? (1.0f - w) : 0.f) + ((hi_i == l) ? w : 0.f);
    accv += wt * v[((size_t)bh * Tt + t) * Dd + d];
    accs += wt;
  }
  Vbar[((size_t)bh * Ll + l) * Dd + d] = accv;
  if (bh == 0 && d == 0) svec[l] = accs;
}

// ---------------------------------------------------------------------------
// 7) out = (est @ Vbar) / (est . svec + eps) + ctx
__global__ void k_final(const float* __restrict__ est, const float* __restrict__ Vbar,
                        const float* __restrict__ svec, const float* __restrict__ ctx,
                        float* __restrict__ out) {
  extern __shared__ float sm[];
  const int tid = threadIdx.x, wv = tid >> 5, lane = tid & 31;
  const int r = lane & 15, hi = lane >> 4;
  float* dbuf = sm + wv * 16;
  const int band = blockIdx.x * (blockDim.x >> 5) + wv;
  const int r0 = band * 16;
  const int bh = r0 / Tt;  // bands never straddle (16 | 2048)
  const float* Vb = Vbar + (size_t)bh * Ll * Dd;
  v8f acc[4];
#pragma unroll
  for (int dt = 0; dt < 4; ++dt) acc[dt] = vzero8();
  for (int k0 = 0; k0 < Ll; k0 += 4) {
    const float* ap = est + (size_t)(r0 + r) * Ll + k0 + hi * 2;
    v2f a; a.x = ap[0]; a.y = ap[1];
#pragma unroll
    for (int dt = 0; dt < 4; ++dt) {
      const float* bp = Vb + (size_t)(k0 + hi * 2) * Dd + dt * 16 + r;
      v2f b; b.x = bp[0]; b.y = bp[Dd];
      acc[dt] = wmma_f32(a, b, acc[dt]);
    }
  }
  if (lane < 16) {
    float s = 0.f;
    const float* erow = est + (size_t)(r0 + lane) * Ll;
    for (int c2 = 0; c2 < Ll; ++c2) s += erow[c2] * svec[c2];
    dbuf[lane] = 1.0f / (s + 1e-6f);
  }
#pragma unroll
  for (int dt = 0; dt < 4; ++dt)
#pragma unroll
    for (int e = 0; e < 8; ++e) {
      size_t idx = (size_t)(r0 + e + hi * 8) * Dd + dt * 16 + r;
      out[idx] = acc[dt][e] * dbuf[e + hi * 8] + ctx[idx];
    }
}

// ---------------------------------------------------------------------------
extern "C" void kernel_launch(void* const* d_in, const int* in_sizes, int n_in,
                              void* d_out, int out_size, void* d_ws, size_t ws_size,
                              hipStream_t stream) {
  (void)in_sizes; (void)n_in; (void)out_size; (void)ws_size;
  const float* q    = (const float*)d_in[0];
  const float* k    = (const float*)d_in[1];
  const float* v    = (const float*)d_in[2];
  const float* proj = (const float*)d_in[3];
  const float* Wenc = (const float*)d_in[4];
  const float* benc = (const float*)d_in[5];
  const float* lng  = (const float*)d_in[6];
  const float* lnb  = (const float*)d_in[7];
  const float* Wdec = (const float*)d_in[8];
  const float* bdec = (const float*)d_in[9];
  float* out = (float*)d_out;
  float* ws  = (float*)d_ws;

  size_t off = 0;
  float* qp    = ws + off; off += (size_t)BHT * Mp;       // 13.37M fl
  float* kpw   = ws + off; off += (size_t)BHT * Mp;       // 13.37M fl
  float* ctx   = ws + off; off += (size_t)BHT * Dd;       //  3.15M fl
  float* est   = ws + off; off += (size_t)BHT * Ll;       //  6.29M fl
  float* Vbar  = ws + off; off += (size_t)BH * Ll * Dd;
  float* svec  = ws + off; off += Ll;
  float* rmq   = ws + off; off += BHT;
  float* rmk   = ws + off; off += BHT;
  float* dgq   = ws + off; off += BHT;
  float* dgk   = ws + off; off += BHT;
  float* qsum  = ws + off; off += BHT;
  float* stabk = ws + off; off += BH;   // total ~36.7M floats (~147 MB)

  const int THR = 256, WPB = 8;  // 8 wave32 per block

  {  // projections
    int blocks = ((BHT / 16) * (Mp / 16)) / WPB;  // 6528
    k_proj<<<blocks, THR, 0, stream>>>(q, proj, qp);
    k_proj<<<blocks, THR, 0, stream>>>(k, proj, kpw);
  }
  {  // stats
    int blocks = BHT / WPB;  // 6144
    k_rowstats<<<blocks, THR, 0, stream>>>(qp, q, rmq, dgq);
    k_rowstats<<<blocks, THR, 0, stream>>>(kpw, k, rmk, dgk);
  }
  k_bhmax<<<BH, 256, 0, stream>>>(rmk, stabk);
  {  // feature exp
    int blocks = BHT / WPB;
    k_featexp<<<blocks, THR, 0, stream>>>(qp, dgq, rmq, stabk, qsum, 1);
    k_featexp<<<blocks, THR, 0, stream>>>(kpw, dgk, rmk, stabk, nullptr, 0);
  }
  {  // causal linear attention: 24 WGs, ~170 KB LDS (320 KB WGP pool)
    size_t smem = (size_t)(Mp * Dd + Mp + 8 * 16 * CH + 8 * 16 + CH * Dd) *
                  sizeof(float);
    k_linattn<<<BH, THR, smem, stream>>>(qp, kpw, v, qsum, ctx);
  }
  {  // predictor
    size_t smem = (size_t)8 * 16 * Ll * sizeof(float);
    k_pred<<<BHT / (16 * WPB), THR, smem, stream>>>(ctx, v, q, Wenc, benc, lng,
                                                    lnb, Wdec, bdec, est);
  }
  k_vbar<<<dim3(Ll, BH), Dd, 0, stream>>>(v, Vbar, svec);
  {  // final combine
    size_t smem = (size_t)8 * 16 * sizeof(float);
    k_final<<<BHT / (16 * WPB), THR, smem, stream>>>(est, Vbar, svec, ctx, out);
  }
}